// GNNAutoEncoder_61366492725622
// MI455X (gfx1250) — compile-verified
//
#include <hip/hip_runtime.h>
#include <hip/hip_bf16.h>

// ---------------------------------------------------------------------------
// GCN autoencoder for MI455X (gfx1250, wave32).
// GEMMs use V_WMMA_F32_16X16X4_F32 (full fp32 matrix pipe, matches reference
// precision). Edge aggregation uses hardware f32 global atomics; all feature
// matrices fit in the 192MB L2 so gather/scatter is on-chip.
// ---------------------------------------------------------------------------

typedef float v2f __attribute__((ext_vector_type(2)));
typedef float v8f __attribute__((ext_vector_type(8)));

__device__ __forceinline__ void atomAddF(float* p, float v) {
#if defined(__HIP_PLATFORM_AMD__)
    unsafeAtomicAdd(p, v);      // global_atomic_add_f32
#else
    atomicAdd(p, v);
#endif
}

// ---------------------------------------------------------------------------
// Degree / normalization
// ---------------------------------------------------------------------------
__global__ __launch_bounds__(256) void k_init_ones(float* __restrict__ p, int n) {
    int i = blockIdx.x * blockDim.x + threadIdx.x;
    if (i < n) p[i] = 1.0f;                       // self-loop contributes +1
}

__global__ __launch_bounds__(256) void k_deg_accum(const int* __restrict__ dst,
                                                   float* __restrict__ deg, int E) {
    int e = blockIdx.x * blockDim.x + threadIdx.x;
    if (e < E) atomAddF(&deg[dst[e]], 1.0f);
}

__global__ __launch_bounds__(256) void k_rsqrt_inplace(float* __restrict__ p, int n) {
    int i = blockIdx.x * blockDim.x + threadIdx.x;
    if (i < n) p[i] = rsqrtf(p[i]);               // dinv = deg^{-1/2}
}

// ---------------------------------------------------------------------------
// H = X @ W  via V_WMMA_F32_16X16X4_F32. One wave per 16x16 output tile.
// A (16x4 f32): lanes 0-15 -> M=lane, K=kk+{0,1}; lanes 16-31 -> K=kk+{2,3}.
// B (4x16 f32): VGPR0 holds K=kk+2*hi row, VGPR1 holds K=kk+2*hi+1 (N=lane&15).
// C/D (16x16 f32): VGPR v -> M = v + 8*(lane>=16), N = lane&15.
// ---------------------------------------------------------------------------
template <int K, int F>
__global__ __launch_bounds__(256) void k_gemm_wmma(const float* __restrict__ X,
                                                   const float* __restrict__ W,
                                                   float* __restrict__ H,
                                                   int tilesM) {
    static_assert(K % 4 == 0 && F % 16 == 0, "shape");
    const int tilesF = F / 16;
    const int wave = blockIdx.x * (blockDim.x >> 5) + (threadIdx.x >> 5);
    if (wave >= tilesM * tilesF) return;          // uniform per-wave guard (EXEC all-1s)
    const int tm = wave / tilesF;
    const int tf = wave % tilesF;
    const int lane = threadIdx.x & 31;
    const int l15  = lane & 15;
    const int hi2  = (lane >> 4) << 1;            // 0 or 2

    const float* xrow = X + (size_t)(tm * 16 + l15) * K;
    const float* wcol = W + (tf * 16 + l15);

    v8f acc = {};
#pragma unroll 4
    for (int kk = 0; kk < K; kk += 4) {
        v2f a, b;
        a.x = xrow[kk + hi2];
        a.y = xrow[kk + hi2 + 1];
        b.x = wcol[(size_t)(kk + hi2) * F];
        b.y = wcol[(size_t)(kk + hi2 + 1) * F];
        acc = __builtin_amdgcn_wmma_f32_16x16x4_f32(
            /*neg_a=*/false, a, /*neg_b=*/false, b,
            /*c_mod=*/(short)0, acc, /*reuse_a=*/false, /*reuse_b=*/false);
    }

    const int hi8 = (lane >> 4) << 3;             // 0 or 8
#pragma unroll
    for (int v = 0; v < 8; ++v)
        H[(size_t)(tm * 16 + hi8 + v) * F + tf * 16 + l15] = acc[v];
}

// Scalar tail for M not divisible by 16 (unused for N=100000, kept for safety).
template <int K, int F>
__global__ __launch_bounds__(256) void k_gemm_tail(const float* __restrict__ X,
                                                   const float* __restrict__ W,
                                                   float* __restrict__ H,
                                                   int rowStart, int rowEnd) {
    int idx = blockIdx.x * blockDim.x + threadIdx.x;
    int rows = rowEnd - rowStart;
    if (idx >= rows * F) return;
    int r = rowStart + idx / F, f = idx % F;
    float s = 0.f;
    for (int k = 0; k < K; ++k) s += X[(size_t)r * K + k] * W[(size_t)k * F + f];
    H[(size_t)r * F + f] = s;
}

// ---------------------------------------------------------------------------
// agg[dst] += h[src] * dinv[src]*dinv[dst]; one thread per (edge, 4-feature chunk)
// consecutive lanes cover consecutive 16B of the same source row -> coalesced.
// ---------------------------------------------------------------------------
template <int F>
__global__ __launch_bounds__(256) void k_scatter(const float* __restrict__ H,
                                                 const int* __restrict__ src,
                                                 const int* __restrict__ dst,
                                                 const float* __restrict__ dinv,
                                                 float* __restrict__ AGG, int E) {
    constexpr int CH = F / 4;
    long long tid = (long long)blockIdx.x * blockDim.x + threadIdx.x;
    if (tid >= (long long)E * CH) return;
    int e = (int)(tid / CH);
    int c = (int)(tid % CH);
    int s = src[e], d = dst[e];
    float w = dinv[s] * dinv[d];
    float4 hv = ((const float4*)(H + (size_t)s * F))[c];
    float* ap = AGG + (size_t)d * F + c * 4;
    atomAddF(ap + 0, hv.x * w);
    atomAddF(ap + 1, hv.y * w);
    atomAddF(ap + 2, hv.z * w);
    atomAddF(ap + 3, hv.w * w);
}

// ---------------------------------------------------------------------------
// out = agg + h * dinv^2 + bias  (+ optional ReLU), in place on AGG
// ---------------------------------------------------------------------------
template <int F, bool RELU>
__global__ __launch_bounds__(256) void k_self_bias(float* __restrict__ AGG,
                                                   const float* __restrict__ H,
                                                   const float* __restrict__ dinv,
                                                   const float* __restrict__ bias,
                                                   int N) {
    long long idx = (long long)blockIdx.x * blockDim.x + threadIdx.x;
    if (idx >= (long long)N * F) return;
    int i = (int)(idx / F), f = (int)(idx % F);
    float ns = dinv[i];
    ns *= ns;
    float v = AGG[idx] + H[idx] * ns + bias[f];
    if (RELU) v = fmaxf(v, 0.0f);
    AGG[idx] = v;
}

// ---------------------------------------------------------------------------
static inline int nblk(long long n, int t) { return (int)((n + t - 1) / t); }

template <int K, int F>
static void launch_gemm(const float* X, const float* W, float* H, int N, hipStream_t s) {
    int tilesM = N / 16;
    long long waves = (long long)tilesM * (F / 16);
    if (waves > 0)
        k_gemm_wmma<K, F><<<nblk(waves, 8), 256, 0, s>>>(X, W, H, tilesM);
    int tail = tilesM * 16;
    if (tail < N)
        k_gemm_tail<K, F><<<nblk((long long)(N - tail) * F, 256), 256, 0, s>>>(X, W, H, tail, N);
}

extern "C" void kernel_launch(void* const* d_in, const int* in_sizes, int n_in,
                              void* d_out, int out_size, void* d_ws, size_t ws_size,
                              hipStream_t stream) {
    const float* x  = (const float*)d_in[0];
    const int*   ei = (const int*)d_in[1];
    const float* W1 = (const float*)d_in[2]; const float* b1 = (const float*)d_in[3];
    const float* W2 = (const float*)d_in[4]; const float* b2 = (const float*)d_in[5];
    const float* W3 = (const float*)d_in[6]; const float* b3 = (const float*)d_in[7];
    const float* W4 = (const float*)d_in[8]; const float* b4 = (const float*)d_in[9];

    const int N = in_sizes[0] / 128;
    const int E = in_sizes[1] / 2;
    const int* src = ei;
    const int* dst = ei + E;

    float* out  = (float*)d_out;
    float* xrec = out;                       // [N,128]
    float* z    = out + (size_t)N * 128;     // [N,32]

    float* ws   = (float*)d_ws;
    float* dinv = ws;                        // [N]
    float* bufH = ws + N;                    // [N,128] GEMM outputs
    float* bufA = bufH + (size_t)N * 128;    // [N,64]  aggregation / next input

    // ---- normalization: deg (incl self loop) -> dinv ----
    k_init_ones<<<nblk(N, 256), 256, 0, stream>>>(dinv, N);
    k_deg_accum<<<nblk(E, 256), 256, 0, stream>>>(dst, dinv, E);
    k_rsqrt_inplace<<<nblk(N, 256), 256, 0, stream>>>(dinv, N);

    // ---- layer 1: 128 -> 64, ReLU ----
    launch_gemm<128, 64>(x, W1, bufH, N, stream);
    hipMemsetAsync(bufA, 0, (size_t)N * 64 * sizeof(float), stream);
    k_scatter<64><<<nblk((long long)E * 16, 256), 256, 0, stream>>>(bufH, src, dst, dinv, bufA, E);
    k_self_bias<64, true><<<nblk((long long)N * 64, 256), 256, 0, stream>>>(bufA, bufH, dinv, b1, N);

    // ---- layer 2: 64 -> 32, latent z (no ReLU) ----
    launch_gemm<64, 32>(bufA, W2, bufH, N, stream);
    hipMemsetAsync(z, 0, (size_t)N * 32 * sizeof(float), stream);
    k_scatter<32><<<nblk((long long)E * 8, 256), 256, 0, stream>>>(bufH, src, dst, dinv, z, E);
    k_self_bias<32, false><<<nblk((long long)N * 32, 256), 256, 0, stream>>>(z, bufH, dinv, b2, N);

    // ---- layer 3: 32 -> 64, ReLU ----
    launch_gemm<32, 64>(z, W3, bufH, N, stream);
    hipMemsetAsync(bufA, 0, (size_t)N * 64 * sizeof(float), stream);
    k_scatter<64><<<nblk((long long)E * 16, 256), 256, 0, stream>>>(bufH, src, dst, dinv, bufA, E);
    k_self_bias<64, true><<<nblk((long long)N * 64, 256), 256, 0, stream>>>(bufA, bufH, dinv, b3, N);

    // ---- layer 4: 64 -> 128, x_rec (no ReLU) ----
    launch_gemm<64, 128>(bufA, W4, bufH, N, stream);
    hipMemsetAsync(xrec, 0, (size_t)N * 128 * sizeof(float), stream);
    k_scatter<128><<<nblk((long long)E * 32, 256), 256, 0, stream>>>(bufH, src, dst, dinv, xrec, E);
    k_self_bias<128, false><<<nblk((long long)N * 128, 256), 256, 0, stream>>>(xrec, bufH, dinv, b4, N);
}